// DifferentiablePointMassSimulator_38671885534076
// MI455X (gfx1250) — compile-verified
//
#include <hip/hip_runtime.h>
#include <cstdint>

// ---------------------------------------------------------------------------
// DifferentiablePointMassSimulator — B=16384 trajectories, H=256 steps, S=8.
// Memory-bound (~168 MB traffic, ~7us floor @ 23.3 TB/s). One thread per
// trajectory. Per-wave double-buffered LDS tiles + CDNA5 async LDS->global
// stores (fully coalesced 512B bursts), pipelined with partial asynccnt waits.
// ---------------------------------------------------------------------------

#define DT_SIM  (1.0f / 30.0f)
#define EPS_SIM 1e-6f

constexpr int kS       = 8;    // state dim
constexpr int kH       = 256;  // horizon
constexpr int kTCh     = 8;    // time steps per chunk
constexpr int kThreads = 128;  // 4 waves / block (leaves room to double-buffer)
constexpr int kWaves   = kThreads / 32;
constexpr int kChunks  = kH / kTCh;

// Per-wave tile: 32 elems x 8 steps x 8 floats = 8 KB. Two buffers per wave.
constexpr int kTileFloats = 32 * kTCh * kS;            // 2048 floats
constexpr int kBufFloats  = kWaves * kTileFloats;      // one buffer set / block

__global__ __launch_bounds__(kThreads)
void pointmass_sim_kernel(const float* __restrict__ init,
                          const float* __restrict__ actions,
                          float* __restrict__ out)
{
    __shared__ float lds[2 * kBufFloats];              // 64 KB

    const int tid  = threadIdx.x;
    const int wave = tid >> 5;
    const int lane = tid & 31;
    const int b    = blockIdx.x * kThreads + tid;

    // Low 32 bits of a generic LDS pointer == DS byte address.
    const uint32_t lds0 = (uint32_t)(uintptr_t)&lds[0];

    // Index (in floats) of this thread's tile within one buffer set.
    const int tileIdx = wave * kTileFloats + lane * (kTCh * kS);

    // Async-store lane mapping: instruction i covers elems {2i, 2i+1};
    // lanes 0..15 -> elem 2i, lanes 16..31 -> elem 2i+1; each lane moves one
    // float4 (16B) of a 256B per-trajectory chunk => 512B contiguous bursts.
    const int e_lane = lane >> 4;
    const int sub16  = lane & 15;

    // LDS byte address of this lane's first burst slot, per buffer.
    const uint32_t lbase0 = lds0 + (uint32_t)(wave * kTileFloats * 4 + lane * 16);
    const uint32_t lbase1 = lbase0 + (uint32_t)(kBufFloats * 4);

    // Global byte address of this lane's first burst slot (chunk 0, i = 0).
    const size_t gelem0 = (size_t)blockIdx.x * kThreads + (size_t)wave * 32 + (size_t)e_lane;
    uint64_t gchunk = (uint64_t)(uintptr_t)out
                    + (gelem0 * kH * kS + (size_t)(sub16 * 4)) * sizeof(float);
    // Strides: +16384 B between store instructions (2 trajectories apart),
    //          +256 B between chunks (kTCh steps * 8 floats * 4 B).
    constexpr uint64_t kGStrideI = 2ull * kH * kS * sizeof(float);      // 16384
    constexpr uint64_t kGStrideC = (uint64_t)kTCh * kS * sizeof(float); // 256

    // Initial state: px,py,vx,vy | extra[4]
    const float4* ist  = reinterpret_cast<const float4*>(init + (size_t)b * kS);
    float4 pv    = ist[0];
    float4 extra = ist[1];
    float px = pv.x, py = pv.y, vx = pv.z, vy = pv.w;

    const float4* act4 = reinterpret_cast<const float4*>(actions + (size_t)b * kH * 2);

    for (int c = 0; c < kChunks; ++c) {
        const int d = c & 1;

        // Buffer d was last used by chunk c-2 (16 stores), chunk c-1 issued 16
        // more. Async stores retire in order: waiting <=16 outstanding
        // guarantees chunk c-2's DMA has drained buffer d, while chunk c-1's
        // stores (other buffer) may still be in flight.
        asm volatile("s_wait_asynccnt 0x10" ::: "memory");

        // ---- actions for 8 steps: 4 x float4 (thrust,torque pairs) ----
        float4 a0 = act4[c * 4 + 0];
        float4 a1 = act4[c * 4 + 1];
        float4 a2 = act4[c * 4 + 2];
        float4 a3 = act4[c * 4 + 3];
        if (c + 1 < kChunks)
            __builtin_prefetch(&act4[(c + 1) * 4], 0, 0);  // global_prefetch_b8

        float th[kTCh], tq[kTCh];
        th[0] = a0.x; tq[0] = a0.y;  th[1] = a0.z; tq[1] = a0.w;
        th[2] = a1.x; tq[2] = a1.y;  th[3] = a1.z; tq[3] = a1.w;
        th[4] = a2.x; tq[4] = a2.y;  th[5] = a2.z; tq[5] = a2.w;
        th[6] = a3.x; tq[6] = a3.y;  th[7] = a3.z; tq[7] = a3.w;

        // Keep the tile access as a direct index into the __shared__ array so
        // clang keeps it in addrspace(3) and emits ds_store_b128 (a generic
        // pointer here regresses to flat stores through the shared aperture).
        float* tile = &lds[(d ? kBufFloats : 0) + tileIdx];

        #pragma unroll
        for (int j = 0; j < kTCh; ++j) {
            float vn  = fmaxf(sqrtf(vx * vx + vy * vy), EPS_SIM);
            float dx  = vx / vn;
            float dy  = vy / vn;
            float ax  = th[j] * dx - tq[j] * dy;   // thrust*dir + torque*perp
            float ay  = th[j] * dy + tq[j] * dx;
            float nvx = vx + ax * DT_SIM;
            float nvy = vy + ay * DT_SIM;
            float npx = px + vx * DT_SIM + 0.5f * ax * (DT_SIM * DT_SIM);
            float npy = py + vy * DT_SIM + 0.5f * ay * (DT_SIM * DT_SIM);
            px = npx; py = npy; vx = nvx; vy = nvy;
            reinterpret_cast<float4*>(tile)[2 * j + 0] = make_float4(px, py, vx, vy);
            reinterpret_cast<float4*>(tile)[2 * j + 1] = extra;  // broadcast extra
        }

        // LDS writes must be complete before the async DMA reads them.
        asm volatile("s_wait_dscnt 0x0" ::: "memory");

        // ---- 16 async bursts from buffer d (each 32 lanes x 16B = 512B) ----
        uint64_t gaddr = gchunk;
        uint32_t loff  = d ? lbase1 : lbase0;
        #pragma unroll
        for (int i = 0; i < 16; ++i) {
            asm volatile("global_store_async_from_lds_b128 %0, %1, off"
                         :: "v"(gaddr), "v"(loff)
                         : "memory");
            gaddr += kGStrideI;   // next pair of trajectories
            loff  += 512;         // next 512B of the tile
        }
        gchunk += kGStrideC;      // advance one chunk of time steps
        // No full drain here: next iteration's partial wait provides the
        // buffer-reuse guarantee; S_ENDPGM performs an implicit wait-idle.
    }
}

extern "C" void kernel_launch(void* const* d_in, const int* in_sizes, int n_in,
                              void* d_out, int out_size, void* d_ws, size_t ws_size,
                              hipStream_t stream) {
    const float* init    = (const float*)d_in[0];   // (B, 8)  fp32
    const float* actions = (const float*)d_in[1];   // (B, 256, 2) fp32
    float*       out     = (float*)d_out;           // (B, 256, 8) fp32

    const int B      = in_sizes[0] / kS;            // 16384
    const int blocks = B / kThreads;                // 128

    pointmass_sim_kernel<<<blocks, kThreads, 0, stream>>>(init, actions, out);
}